// UCCAEncoder_13280038879907
// MI455X (gfx1250) — compile-verified
//
#include <hip/hip_runtime.h>
#include <math.h>
#include <stdint.h>

// ---------------------------------------------------------------------------
// UCCA GAT encoder forward for MI455X (gfx1250, wave32, WMMA).
// GEMMs: V_WMMA_F32_16X16X4_F32 (exact fp32), block = 8 waves x (16 x 16*NT),
// B tile double-buffered in LDS via global_load_async_to_lds_b128 (ASYNCcnt).
// Segment softmax / LN / GRU / scatter are VALU + atomics (bandwidth-trivial
// next to ~790 GFLOP of matrix work).
// ---------------------------------------------------------------------------

#define H_       8
#define DH_      32
#define DIM_     256
#define FFNH_    2048
#define NLAYERS_ 3
#define NITER_   3
#define LRELU_   0.2f
#define KSCALE_  0.17677669529663687f   // 32^-0.5

#define USE_ASYNC_STAGE 1

typedef __attribute__((ext_vector_type(2))) float v2f;
typedef __attribute__((ext_vector_type(8))) float v8f;

// ------------------------------- helpers -----------------------------------

__device__ __forceinline__ float warp_sum(float v) {
#pragma unroll
  for (int off = 16; off; off >>= 1) v += __shfl_xor(v, off, 32);
  return v;
}

__device__ __forceinline__ uint32_t hashu(uint32_t x) {
  x ^= x >> 16; x *= 0x7feb352dU;
  x ^= x >> 15; x *= 0x846ca68bU;
  x ^= x >> 16; return x;
}

__device__ float atomicMaxF(float* addr, float val) {
  int* ai = (int*)addr;
  int old = *ai;
  while (__int_as_float(old) < val) {
    int prev = atomicCAS(ai, old, __float_as_int(val));
    if (prev == old) break;
    old = prev;
  }
  return __int_as_float(old);
}

// ------------------------------- WMMA GEMM ---------------------------------
// C[M,N] = act(scale * (A[M,K] @ B[K,N]) (+ C if ACC) (+ bias if BIAS))
// Row-major, lda=K, ldb=N, ldc=N.  K % 32 == 0, N % (16*NT) == 0.
// Block: 256 threads = 8 waves stacked along M (block tile 128 x 16*NT).
// Each wave: one 16-row strip, NT accumulators of 16x16 (A fragment reused NT
// times per K-step). B K-tiles (32 x 16*NT) double-buffered in LDS, filled by
// global_load_async_to_lds_b128; next tile staged while current one computes.
// A 16x4 fp32 fragment: lanes 0-15 hold (k,k+1) of row M=lane; lanes 16-31
// hold (k+2,k+3) of row M=lane-16. B mirrored over N. C/D: VGPR r -> row r
// (lanes 0-15) / row r+8 (lanes 16-31), col = lane&15.

template <int NW>
__device__ __forceinline__ void stage_tile(const float* __restrict__ B, int N,
                                           int k0, int n0, float* lds_dst,
                                           uint32_t lds_off) {
#pragma unroll
  for (int u = 0; u < (32 * NW) / (256 * 4); ++u) {
    const int idx = ((int)threadIdx.x + u * 256) * 4;
    const int kk = idx / NW, nn = idx % NW;
    const float* gp = B + (size_t)(k0 + kk) * N + n0 + nn;
#if USE_ASYNC_STAGE
    // LDS byte address: shared block is the only LDS object -> base 0.
    const uint32_t laddr = lds_off + (uint32_t)(idx * 4);
    asm volatile("global_load_async_to_lds_b128 %0, %1, off"
                 :: "v"(laddr), "v"(gp) : "memory");
#else
    *(float4*)&lds_dst[idx] = *(const float4*)gp;
#endif
  }
}

__device__ __forceinline__ void stage_fence() {
#if USE_ASYNC_STAGE
  asm volatile("s_wait_asynccnt 0x0" ::: "memory");
#endif
  __syncthreads();
}

template <int ACT, bool ACC, bool BIAS, int NT>
__global__ __launch_bounds__(256) void gemm_wmma(
    const float* __restrict__ A, const float* __restrict__ B,
    const float* __restrict__ bias, float* __restrict__ C,
    int M, int N, int K, float scale) {
  constexpr int NW = 16 * NT;                    // block N width
  constexpr uint32_t TILEB = 32u * NW * 4u;      // tile bytes
  __shared__ __align__(16) float Bs[2][32 * NW]; // double-buffered B tiles

  const int lane = threadIdx.x & 31;
  const int wave = threadIdx.x >> 5;
  int m0 = (blockIdx.y * 8 + wave) * 16;
  const int n0 = blockIdx.x * NW;
  const bool active = (m0 < M);
  if (m0 + 16 > M) m0 = M - 16;   // clamp: keep all waves in the barriers
  const int half = lane >> 4;     // 0 | 1
  const int l15  = lane & 15;
  const int mb   = m0 + half * 8;

  v8f acc[NT];
#pragma unroll
  for (int nt = 0; nt < NT; ++nt) {
    if (ACC) {
#pragma unroll
      for (int r = 0; r < 8; ++r)
        acc[nt][r] = C[(size_t)(mb + r) * N + n0 + nt * 16 + l15];
    } else {
      acc[nt] = (v8f){0.f, 0.f, 0.f, 0.f, 0.f, 0.f, 0.f, 0.f};
    }
  }

  const float* Arow = A + (size_t)(m0 + l15) * K;

  // prologue: stage first K-tile into buffer 0
  stage_tile<NW>(B, N, 0, n0, Bs[0], 0);
  stage_fence();

  int buf = 0;
  for (int k0 = 0; k0 < K; k0 += 32, buf ^= 1) {
    // stage next tile into the other buffer (overlaps with compute below)
    if (k0 + 32 < K)
      stage_tile<NW>(B, N, k0 + 32, n0, Bs[buf ^ 1], (buf ^ 1) ? TILEB : 0u);

    const float* Bc = Bs[buf];
#pragma unroll
    for (int kk = 0; kk < 32; kk += 4) {
      const int ka = k0 + kk + half * 2;
      v2f a;
      a.x = Arow[ka];
      a.y = Arow[ka + 1];
      const int kb = kk + half * 2;
#pragma unroll
      for (int nt = 0; nt < NT; ++nt) {
        v2f b;
        b.x = Bc[kb * NW + nt * 16 + l15];
        b.y = Bc[(kb + 1) * NW + nt * 16 + l15];
        acc[nt] = __builtin_amdgcn_wmma_f32_16x16x4_f32(
            false, a, false, b, (short)0, acc[nt], false, false);
      }
    }
    // my async stage done + everyone finished reading current buffer
    stage_fence();
  }

  if (active) {
#pragma unroll
    for (int nt = 0; nt < NT; ++nt) {
      const int cn = n0 + nt * 16 + l15;
      const float bv = BIAS ? bias[cn] : 0.f;
#pragma unroll
      for (int r = 0; r < 8; ++r) {
        float v = acc[nt][r] * scale + bv;
        if (ACT == 1) v = fmaxf(v, 0.f);
        C[(size_t)(mb + r) * N + cn] = v;
      }
    }
  }
}

template <int ACT, bool ACC, bool BIAS>
static void launch_gemm_t(const float* A, const float* B, const float* bias,
                          float* C, int M, int N, int K, float scale,
                          hipStream_t s) {
  if (N % 64 == 0) {
    dim3 g(N / 64, (M + 127) / 128), b(256);
    gemm_wmma<ACT, ACC, BIAS, 4><<<g, b, 0, s>>>(A, B, bias, C, M, N, K, scale);
  } else {  // N == 32 cases
    dim3 g(N / 32, (M + 127) / 128), b(256);
    gemm_wmma<ACT, ACC, BIAS, 2><<<g, b, 0, s>>>(A, B, bias, C, M, N, K, scale);
  }
}

static void launch_gemm(const float* A, const float* B, const float* bias,
                        float* C, int M, int N, int K, float scale,
                        int act, int acc, int has_bias, hipStream_t s) {
  if (acc)            launch_gemm_t<1, true,  true >(A, B, bias, C, M, N, K, scale, s);
  else if (has_bias) {
    if (act)          launch_gemm_t<1, false, true >(A, B, bias, C, M, N, K, scale, s);
    else              launch_gemm_t<0, false, true >(A, B, bias, C, M, N, K, scale, s);
  } else              launch_gemm_t<0, false, false>(A, B, bias, C, M, N, K, scale, s);
}

// --------------------------- elementwise kernels ---------------------------

__global__ void k_fill(float* p, float v, size_t n) {
  size_t i = (size_t)blockIdx.x * blockDim.x + threadIdx.x;
  if (i < n) p[i] = v;
}

__global__ void k_gather_label(const int* __restrict__ lab,
                               const float* __restrict__ emb,
                               float* __restrict__ out, int E) {
  size_t i = (size_t)blockIdx.x * blockDim.x + threadIdx.x;
  if (i >= (size_t)E * DIM_) return;
  int e = (int)(i >> 8), j = (int)(i & 255);
  out[i] = emb[(size_t)lab[e] * DIM_ + j];
}

// x_cat[e][h][0:32] = h[dst[e]][h*32+...] ; x_cat[e][h][32:64] = h[src[e]][...]
__global__ void k_gather_xcat(const float* __restrict__ hbuf,
                              const int* __restrict__ srcv,
                              const int* __restrict__ dstv,
                              float* __restrict__ xcat, int E) {
  size_t i = (size_t)blockIdx.x * blockDim.x + threadIdx.x;
  if (i >= (size_t)E * 512) return;
  int e = (int)(i >> 9), r = (int)(i & 511);
  int hh = r >> 6, c = r & 63;
  int node = (c < 32) ? dstv[e] : srcv[e];
  int d = (c < 32) ? c : (c - 32);
  xcat[i] = hbuf[(size_t)node * DIM_ + hh * 32 + d];
}

// slots = mu[d] + exp(log_sigma[d]) * N(0,1)  (hash RNG; JAX threefry not
// bit-reproducible here — distributional match only)
__global__ void k_init_slots(float* __restrict__ slots,
                             const float* __restrict__ mu,
                             const float* __restrict__ lsig,
                             size_t n, uint32_t seed) {
  size_t i = (size_t)blockIdx.x * blockDim.x + threadIdx.x;
  if (i >= n) return;
  uint32_t h1 = hashu(seed ^ (uint32_t)i);
  uint32_t h2 = hashu(h1 + 0x9e3779b9u);
  float u1 = ((h1 >> 8) + 1u) * (1.0f / 16777216.0f);
  float u2 = (h2 >> 8) * (1.0f / 16777216.0f);
  float nrm = sqrtf(-2.0f * logf(u1)) * __cosf(6.283185307179586f * u2);
  int d = (int)(i & 31);
  slots[i] = mu[d] + expf(lsig[d]) * nrm;
}

// LayerNorm over last dim (32) of [rows, 32]; one wave per row.
__global__ void k_ln_head(const float* __restrict__ x, const float* __restrict__ g,
                          const float* __restrict__ b, float* __restrict__ y,
                          int rows) {
  int row = blockIdx.x * 8 + (threadIdx.x >> 5);
  int d = threadIdx.x & 31;
  if (row >= rows) return;
  float v = x[(size_t)row * 32 + d];
  float m = warp_sum(v) * (1.0f / 32.0f);
  float c0 = v - m;
  float var = warp_sum(c0 * c0) * (1.0f / 32.0f);
  y[(size_t)row * 32 + d] = c0 * rsqrtf(var + 1e-5f) * g[d] + b[d];
}

// out[e,h] = leaky_relu( dot(X[e,h,:per], w[h,:per]) )
__global__ void k_head_logits(const float* __restrict__ X,
                              const float* __restrict__ w,
                              float* __restrict__ out, int E, int per) {
  int i = blockIdx.x * blockDim.x + threadIdx.x;
  if (i >= E * H_) return;
  int hh = i & 7;
  const float* xr = X + (size_t)i * per;
  const float* wr = w + (size_t)hh * per;
  float s = 0.f;
  for (int c = 0; c < per; ++c) s += xr[c] * wr[c];
  out[i] = (s < 0.f) ? LRELU_ * s : s;
}

__global__ void k_segmax(const float* __restrict__ logit, const int* __restrict__ dst,
                         float* __restrict__ nmax, int E) {
  int i = blockIdx.x * blockDim.x + threadIdx.x;
  if (i >= E * H_) return;
  atomicMaxF(&nmax[(size_t)dst[i >> 3] * H_ + (i & 7)], logit[i]);
}

__global__ void k_segexp(float* __restrict__ logit, const int* __restrict__ dst,
                         const float* __restrict__ nmax, float* __restrict__ nsum,
                         int E) {
  int i = blockIdx.x * blockDim.x + threadIdx.x;
  if (i >= E * H_) return;
  size_t s = (size_t)dst[i >> 3] * H_ + (i & 7);
  float v = expf(logit[i] - nmax[s]);
  logit[i] = v;
  atomicAdd(&nsum[s], v);
}

__global__ void k_segnorm(float* __restrict__ logit, const int* __restrict__ dst,
                          const float* __restrict__ nsum, int E) {
  int i = blockIdx.x * blockDim.x + threadIdx.x;
  if (i >= E * H_) return;
  logit[i] = logit[i] / (nsum[(size_t)dst[i >> 3] * H_ + (i & 7)] + 1e-16f);
}

// upd = k * alpha broadcast over DH
__global__ void k_upd(const float* __restrict__ kmat, const float* __restrict__ alpha,
                      float* __restrict__ upd, int E) {
  size_t i = (size_t)blockIdx.x * blockDim.x + threadIdx.x;
  if (i >= (size_t)E * DIM_) return;
  int e = (int)(i >> 8), hh = (int)((i >> 5) & 7);
  upd[i] = kmat[i] * alpha[(size_t)e * H_ + hh];
}

// GRUCell gates (gx, gh precomputed = u@Wi+bi, h@Wh+bh), layout [rows,768]
__global__ void k_gru(const float* __restrict__ gx, const float* __restrict__ gh,
                      const float* __restrict__ hprev, float* __restrict__ hout,
                      int rows) {
  size_t i = (size_t)blockIdx.x * blockDim.x + threadIdx.x;
  if (i >= (size_t)rows * DIM_) return;
  int r = (int)(i >> 8), j = (int)(i & 255);
  size_t b0 = (size_t)r * 768 + j;
  float xr = gx[b0], xz = gx[b0 + 256], xn = gx[b0 + 512];
  float hr = gh[b0], hz = gh[b0 + 256], hn = gh[b0 + 512];
  float rr = 1.f / (1.f + expf(-(xr + hr)));
  float zz = 1.f / (1.f + expf(-(xz + hz)));
  float nn = tanhf(xn + rr * hn);
  hout[i] = (1.f - zz) * nn + zz * hprev[i];
}

__global__ void k_add(float* __restrict__ y, const float* __restrict__ a,
                      const float* __restrict__ b, size_t n) {
  size_t i = (size_t)blockIdx.x * blockDim.x + threadIdx.x;
  if (i < n) y[i] = a[i] + b[i];
}

// g[e] = sigmoid(la[e,:]·w1 + xlab[e,:]·w2); one wave per edge
__global__ void k_gate(const float* __restrict__ la, const float* __restrict__ xlab,
                       const float* __restrict__ w1, const float* __restrict__ w2,
                       float* __restrict__ g, int E) {
  int e = blockIdx.x * 8 + (threadIdx.x >> 5);
  int lane = threadIdx.x & 31;
  if (e >= E) return;
  float s = 0.f;
  for (int c = lane; c < DIM_; c += 32)
    s += la[(size_t)e * DIM_ + c] * w1[c] + xlab[(size_t)e * DIM_ + c] * w2[c];
  s = warp_sum(s);
  if (lane == 0) g[e] = 1.f / (1.f + expf(-s));
}

__global__ void k_gres(const float* __restrict__ g, const float* __restrict__ la,
                       const float* __restrict__ xlab, float* __restrict__ xl, int E) {
  size_t i = (size_t)blockIdx.x * blockDim.x + threadIdx.x;
  if (i >= (size_t)E * DIM_) return;
  float gv = g[i >> 8];
  xl[i] = gv * la[i] + (1.f - gv) * xlab[i];
}

// ef[e, h*32+d] = xj * alpha  (xj = second half of x_cat head block)
__global__ void k_ef(const float* __restrict__ xcat, const float* __restrict__ alpha,
                     float* __restrict__ ef, int E) {
  size_t i = (size_t)blockIdx.x * blockDim.x + threadIdx.x;
  if (i >= (size_t)E * DIM_) return;
  int e = (int)(i >> 8), hh = (int)((i >> 5) & 7), d = (int)(i & 31);
  ef[i] = xcat[(size_t)e * 512 + hh * 64 + 32 + d] * alpha[(size_t)e * H_ + hh];
}

// msg = ef + ffn_out; scatter-add into x_next[dst]
__global__ void k_msg_scatter(const float* __restrict__ ef, const float* __restrict__ ffo,
                              const int* __restrict__ dst, float* __restrict__ xnext,
                              int e0, int rows) {
  size_t i = (size_t)blockIdx.x * blockDim.x + threadIdx.x;
  if (i >= (size_t)rows * DIM_) return;
  int r = (int)(i >> 8), j = (int)(i & 255);
  int e = e0 + r;
  float v = ef[(size_t)e * DIM_ + j] + ffo[i];
  atomicAdd(&xnext[(size_t)dst[e] * DIM_ + j], v);
}

// ------------------------------- host side ---------------------------------

static inline dim3 g1(size_t n) { return dim3((unsigned)((n + 255) / 256)); }

extern "C" void kernel_launch(void* const* d_in, const int* in_sizes, int n_in,
                              void* d_out, int out_size, void* d_ws, size_t ws_size,
                              hipStream_t stream) {
  const float* x_in      = (const float*)d_in[0];
  const int*   eidx      = (const int*)d_in[1];
  const int*   elab      = (const int*)d_in[2];
  const float* label_emb = (const float*)d_in[3];
  const int E  = in_sizes[2];
  const int NN = in_sizes[0] / DIM_;
  const int* srcv = eidx;          // edge_index[0] = source j
  const int* dstv = eidx + E;      // edge_index[1] = target i

  // ---- workspace carve (256B aligned) ----
  char* wp = (char*)d_ws;
  auto alloc = [&](size_t nfl) -> float* {
    float* p = (float*)wp;
    wp += ((nfl * sizeof(float) + 255) / 256) * 256;
    return p;
  };
  const int CH  = 8192;   // GRU chunk (edges)
  const int CHC = 4096;   // combine-FFN chunk (edges)
  const int CHM = 16384;  // slot-MLP chunk (E*H rows)

  float* xlab  = alloc((size_t)E * DIM_);
  float* hbuf  = alloc((size_t)NN * DIM_);
  float* xcat  = alloc((size_t)E * 512);
  float* kmat  = alloc((size_t)E * DIM_);
  float* slots = alloc((size_t)E * DIM_);
  float* sprev = alloc((size_t)E * DIM_);
  float* sbuf  = alloc((size_t)E * DIM_);
  float* qbuf  = alloc((size_t)E * DIM_);
  float* logit = alloc((size_t)E * H_);
  float* nmax  = alloc((size_t)NN * H_);
  float* nsum  = alloc((size_t)NN * H_);
  float* upd   = alloc((size_t)E * DIM_);
  float* xl    = alloc((size_t)E * DIM_);
  float* ef    = alloc((size_t)E * DIM_);
  float* eo    = alloc((size_t)E * DIM_);
  float* nodeA = alloc((size_t)NN * DIM_);
  float* nodeB = alloc((size_t)NN * DIM_);
  float* gxc   = alloc((size_t)CH * 768);
  float* ghc   = alloc((size_t)CH * 768);
  float* hidc  = alloc((size_t)CHC * FFNH_);   // also reused for slot-MLP hidden
  float* ffoc  = alloc((size_t)CHC * DIM_);    // also reused for slot-MLP out
  float* gbuf  = alloc((size_t)E);
  (void)ws_size; (void)n_in; (void)out_size;

  // x_label (same for all layers)
  k_gather_label<<<g1((size_t)E * DIM_), 256, 0, stream>>>(elab, label_emb, xlab, E);

  for (int l = 0; l < NLAYERS_; ++l) {
    const float* x_cur = (l == 0) ? x_in : (l == 1 ? nodeA : nodeB);
    float* x_next = (l == 0) ? nodeA : (l == 1 ? nodeB : (float*)d_out);

    // ---- layer params (JAX pytree: dict keys in sorted order) ----
    const int pb = 4 + l * 27;
    const float* att    = (const float*)d_in[pb + 0];   // [8,64]
    const float* b_lin  = (const float*)d_in[pb + 1];
    const float* c_b1   = (const float*)d_in[pb + 2];
    const float* c_b2   = (const float*)d_in[pb + 3];
    const float* c_w1   = (const float*)d_in[pb + 4];   // [512,2048]
    const float* c_w2   = (const float*)d_in[pb + 5];   // [2048,256]
    const float* g_w1   = (const float*)d_in[pb + 6];
    const float* g_w2   = (const float*)d_in[pb + 7];
    const float* s_att  = (const float*)d_in[pb + 8];   // [8,32]
    const float* gru_bh = (const float*)d_in[pb + 9];
    const float* gru_bi = (const float*)d_in[pb + 10];
    const float* gru_wh = (const float*)d_in[pb + 11];  // [256,768]
    const float* gru_wi = (const float*)d_in[pb + 12];
    const float* lsig   = (const float*)d_in[pb + 13];
    const float* m_b1   = (const float*)d_in[pb + 14];
    const float* m_b2   = (const float*)d_in[pb + 15];
    const float* m_w1   = (const float*)d_in[pb + 16];  // [32,256]
    const float* m_w2   = (const float*)d_in[pb + 17];  // [256,32]
    const float* mu     = (const float*)d_in[pb + 18];
    const float* nm_b   = (const float*)d_in[pb + 19];
    const float* nm_g   = (const float*)d_in[pb + 20];
    const float* ns_b   = (const float*)d_in[pb + 21];
    const float* ns_g   = (const float*)d_in[pb + 22];
    const float* wk     = (const float*)d_in[pb + 23];  // [64,32]
    const float* wq     = (const float*)d_in[pb + 24];  // [32,32]
    const float* w_lab  = (const float*)d_in[pb + 25];  // [256,256]
    const float* w_lin  = (const float*)d_in[pb + 26];  // [256,256]

    // h = x @ w_lin + b_lin
    launch_gemm(x_cur, w_lin, b_lin, hbuf, NN, DIM_, DIM_, 1.f, 0, 0, 1, stream);
    // x_cat gather
    k_gather_xcat<<<g1((size_t)E * 512), 256, 0, stream>>>(hbuf, srcv, dstv, xcat, E);
    // k = (x_cat @ wk) * DH^-0.5    [E*H,64]@[64,32]
    launch_gemm(xcat, wk, nullptr, kmat, E * H_, DH_, 2 * DH_, KSCALE_, 0, 0, 0, stream);
    // slots init
    k_init_slots<<<g1((size_t)E * DIM_), 256, 0, stream>>>(
        slots, mu, lsig, (size_t)E * DIM_, 0x2au * 2654435761u + (uint32_t)l);

    for (int it = 0; it < NITER_; ++it) {
      hipMemcpyAsync(sprev, slots, (size_t)E * DIM_ * 4, hipMemcpyDeviceToDevice, stream);
      // s = LN(slots); q = (s @ wq) * DH^-0.5
      k_ln_head<<<(E * H_ + 7) / 8, 256, 0, stream>>>(slots, ns_g, ns_b, sbuf, E * H_);
      launch_gemm(sbuf, wq, nullptr, qbuf, E * H_, DH_, DH_, KSCALE_, 0, 0, 0, stream);
      // a = lrelu(q · atten); segment softmax over dst
      k_head_logits<<<g1((size_t)E * H_), 256, 0, stream>>>(qbuf, s_att, logit, E, DH_);
      k_fill<<<g1((size_t)NN * H_), 256, 0, stream>>>(nmax, -3.4e38f, (size_t)NN * H_);
      k_fill<<<g1((size_t)NN * H_), 256, 0, stream>>>(nsum, 0.f, (size_t)NN * H_);
      k_segmax<<<g1((size_t)E * H_), 256, 0, stream>>>(logit, dstv, nmax, E);
      k_segexp<<<g1((size_t)E * H_), 256, 0, stream>>>(logit, dstv, nmax, nsum, E);
      k_segnorm<<<g1((size_t)E * H_), 256, 0, stream>>>(logit, dstv, nsum, E);
      // upd = k * a ; GRU (chunked over edges)
      k_upd<<<g1((size_t)E * DIM_), 256, 0, stream>>>(kmat, logit, upd, E);
      for (int e0 = 0; e0 < E; e0 += CH) {
        int rows = (E - e0 < CH) ? (E - e0) : CH;
        launch_gemm(upd + (size_t)e0 * DIM_, gru_wi, gru_bi, gxc, rows, 768, DIM_, 1.f, 0, 0, 1, stream);
        launch_gemm(sprev + (size_t)e0 * DIM_, gru_wh, gru_bh, ghc, rows, 768, DIM_, 1.f, 0, 0, 1, stream);
        k_gru<<<g1((size_t)rows * DIM_), 256, 0, stream>>>(
            gxc, ghc, sprev + (size_t)e0 * DIM_, slots + (size_t)e0 * DIM_, rows);
      }
      // slots += ffn(LN(slots)) per head (chunked over E*H rows)
      k_ln_head<<<(E * H_ + 7) / 8, 256, 0, stream>>>(slots, nm_g, nm_b, sbuf, E * H_);
      for (int r0 = 0; r0 < E * H_; r0 += CHM) {
        int rows = (E * H_ - r0 < CHM) ? (E * H_ - r0) : CHM;
        launch_gemm(sbuf + (size_t)r0 * DH_, m_w1, m_b1, hidc, rows, 256, DH_, 1.f, 1, 0, 1, stream);
        launch_gemm(hidc, m_w2, m_b2, ffoc, rows, DH_, 256, 1.f, 0, 0, 1, stream);
        k_add<<<g1((size_t)rows * DH_), 256, 0, stream>>>(
            slots + (size_t)r0 * DH_, slots + (size_t)r0 * DH_, ffoc, (size_t)rows * DH_);
      }
    }

    // gating residual: xl = g*la + (1-g)*x_label  (la = slots)
    k_gate<<<(E + 7) / 8, 256, 0, stream>>>(slots, xlab, g_w1, g_w2, gbuf, E);
    k_gres<<<g1((size_t)E * DIM_), 256, 0, stream>>>(gbuf, slots, xlab, xl, E);

    // alpha = seg_softmax(lrelu(x_cat · att))
    k_head_logits<<<g1((size_t)E * H_), 256, 0, stream>>>(xcat, att, logit, E, 2 * DH_);
    k_fill<<<g1((size_t)NN * H_), 256, 0, stream>>>(nmax, -3.4e38f, (size_t)NN * H_);
    k_fill<<<g1((size_t)NN * H_), 256, 0, stream>>>(nsum, 0.f, (size_t)NN * H_);
    k_segmax<<<g1((size_t)E * H_), 256, 0, stream>>>(logit, dstv, nmax, E);
    k_segexp<<<g1((size_t)E * H_), 256, 0, stream>>>(logit, dstv, nmax, nsum, E);
    k_segnorm<<<g1((size_t)E * H_), 256, 0, stream>>>(logit, dstv, nsum, E);

    // ef = xj * alpha ; eo = ef @ w_label + xl
    k_ef<<<g1((size_t)E * DIM_), 256, 0, stream>>>(xcat, logit, ef, E);
    launch_gemm(ef, w_lab, nullptr, eo, E, DIM_, DIM_, 1.f, 0, 0, 0, stream);
    k_add<<<g1((size_t)E * DIM_), 256, 0, stream>>>(eo, eo, xl, (size_t)E * DIM_);

    // x_next = segment_sum(ef + ffn([ef,eo]))   (combine FFN chunked)
    k_fill<<<g1((size_t)NN * DIM_), 256, 0, stream>>>(x_next, 0.f, (size_t)NN * DIM_);
    for (int e0 = 0; e0 < E; e0 += CHC) {
      int rows = (E - e0 < CHC) ? (E - e0) : CHC;
      // hid = ef@W_top ; hid = relu(hid + eo@W_bot + b1)
      launch_gemm(ef + (size_t)e0 * DIM_, c_w1, nullptr, hidc, rows, FFNH_, DIM_, 1.f, 0, 0, 0, stream);
      launch_gemm(eo + (size_t)e0 * DIM_, c_w1 + (size_t)DIM_ * FFNH_, c_b1, hidc, rows, FFNH_, DIM_, 1.f, 1, 1, 1, stream);
      // out = hid @ c_w2 + b2
      launch_gemm(hidc, c_w2, c_b2, ffoc, rows, DIM_, FFNH_, 1.f, 0, 0, 1, stream);
      k_msg_scatter<<<g1((size_t)rows * DIM_), 256, 0, stream>>>(ef, ffoc, dstv, x_next, e0, rows);
    }
  }
}